// ApplyCoeffs_38044820308636
// MI455X (gfx1250) — compile-verified
//
#include <hip/hip_runtime.h>
#include <stdint.h>

// ---------------------------------------------------------------------------
// ApplyCoeffs (bilateral-grid style, CG==1 degenerate case):
//   s        = sum_c input[n,c,h,w]
//   out[n,g] = coeff[n,2g,h,w] * s + coeff[n,2g+1,h,w]
// Pure streaming kernel: 128 B HBM traffic / pixel, ~23 flops / pixel.
// ~1.07 GB total traffic -> ~46 us floor at 23.3 TB/s. 20x below the ridge
// point, so the MI455X-specific lever is the memory path: b128 accesses,
// async global->LDS double-buffered prefetch, NT hints on stream-once data.
// ---------------------------------------------------------------------------

typedef float v4f __attribute__((ext_vector_type(4)));
typedef int   v4i __attribute__((vector_size(16)));   // matches builtin's V4i

#define AS1 __attribute__((address_space(1)))
#define AS3 __attribute__((address_space(3)))

#if defined(__HIP_DEVICE_COMPILE__)
  #if __has_builtin(__builtin_amdgcn_global_load_async_to_lds_b128)
    #define HAVE_ASYNC_LDS 1
  #else
    #define HAVE_ASYNC_LDS 0
    #warning "gfx1250: __builtin_amdgcn_global_load_async_to_lds_b128 unavailable; using direct loads"
  #endif
  #if __has_builtin(__builtin_amdgcn_s_wait_asynccnt)
    #define WAIT_ASYNC(n) __builtin_amdgcn_s_wait_asynccnt(n)
  #else
    #define WAIT_ASYNC(n) asm volatile("s_wait_asynccnt " #n ::: "memory")
  #endif
#else
  #define HAVE_ASYNC_LDS 0
  #define WAIT_ASYNC(n)
#endif

namespace {
constexpr int       kG     = 8;
constexpr int       kC     = 8;
constexpr int       kN     = 4;
constexpr long long kHW    = 1024LL * 2048LL;      // pixels per plane
constexpr int       kBlock = 256;                  // 8 waves (wave32)
constexpr int       kPPT   = 4;                    // pixels per thread (float4)
constexpr int       kTile  = kBlock * kPPT;        // 1024 pixels per tile
constexpr long long kTotalPix   = (long long)kN * kHW;          // 8,388,608
constexpr int       kTotalTiles = (int)(kTotalPix / kTile);     // 8192
constexpr int       kGrid       = 2048;            // 4 tiles per block
} // namespace

__global__ __launch_bounds__(kBlock) void ApplyCoeffs_38044820308636_kernel(
    const float* __restrict__ coeff,   // (N, 16, H, W)
    const float* __restrict__ in,      // (N,  8, H, W)
    float* __restrict__ out)           // (N,  8, H, W)
{
    // Double-buffered input staging: 2 * 8 planes * 1024 px * 4 B = 64 KB LDS
    // -> 5 blocks (40 waves) per WGP against the 320 KB pool.
    __shared__ __align__(16) float lds[2][kC][kTile];

    const int tid = (int)threadIdx.x;

    // Issue 8 async b128 copies (one per input channel plane) for one tile.
    auto issue = [&](int tile, int buf) {
        const long long pixbase = (long long)tile * kTile;
        const int       n  = (int)(pixbase / kHW);
        const long long p  = pixbase - (long long)n * kHW + (long long)(tid * kPPT);
        const float*    g  = in + (long long)n * kC * kHW + p;
        float*          l  = &lds[buf][0][tid * kPPT];
#if HAVE_ASYNC_LDS
        #pragma unroll
        for (int c = 0; c < kC; ++c) {
            __builtin_amdgcn_global_load_async_to_lds_b128(
                (AS1 v4i*)(g + (long long)c * kHW),
                (AS3 v4i*)(l + c * kTile),
                /*offset=*/0, /*cpol=*/0);
        }
#else
        #pragma unroll
        for (int c = 0; c < kC; ++c) {
            *(v4f*)(l + c * kTile) = *(const v4f*)(g + (long long)c * kHW);
        }
#endif
    };

    int tile = (int)blockIdx.x;
    int buf  = 0;
    issue(tile, buf);

    for (; tile < kTotalTiles; tile += kGrid) {
        const int next = tile + kGrid;
        const bool have_next = (next < kTotalTiles);
        if (have_next) issue(next, buf ^ 1);

#if HAVE_ASYNC_LDS
        if (have_next) { WAIT_ASYNC(8); }   // current tile's 8 copies done
        else           { WAIT_ASYNC(0); }
#endif

        const long long pixbase = (long long)tile * kTile;
        const int       n  = (int)(pixbase / kHW);
        const long long p  = pixbase - (long long)n * kHW + (long long)(tid * kPPT);

        // s = sum over 8 channels (read back the staged planes: ds_load_b128)
        v4f s = *(const v4f*)&lds[buf][0][tid * kPPT];
        #pragma unroll
        for (int c = 1; c < kC; ++c) {
            s += *(const v4f*)&lds[buf][c][tid * kPPT];
        }

        const float* cbase = coeff + (long long)n * (2 * kG) * kHW + p;
        float*       obase = out   + (long long)n * kG       * kHW + p;

        #pragma unroll
        for (int g = 0; g < kG; ++g) {
            v4f A = __builtin_nontemporal_load((const v4f*)(cbase + (long long)(2 * g)     * kHW));
            v4f B = __builtin_nontemporal_load((const v4f*)(cbase + (long long)(2 * g + 1) * kHW));
            v4f r = A * s + B;
            __builtin_nontemporal_store(r, (v4f*)(obase + (long long)g * kHW));
        }

        buf ^= 1;
    }
}

extern "C" void kernel_launch(void* const* d_in, const int* in_sizes, int n_in,
                              void* d_out, int out_size, void* d_ws, size_t ws_size,
                              hipStream_t stream) {
    (void)in_sizes; (void)n_in; (void)out_size; (void)d_ws; (void)ws_size;
    const float* coeff = (const float*)d_in[0];   // (4, 16, 1024, 2048) f32
    const float* input = (const float*)d_in[1];   // (4,  8, 1024, 2048) f32
    float*       out   = (float*)d_out;           // (4,  8, 1024, 2048) f32
    ApplyCoeffs_38044820308636_kernel<<<kGrid, kBlock, 0, stream>>>(coeff, input, out);
}